// MultivariateMuyGPs_layer_27015344292374
// MI455X (gfx1250) — compile-verified
//
#include <hip/hip_runtime.h>
#include <math.h>

// MuyGPs layer for MI455X (gfx1250, wave32).
// Per-batch gram (xn·xn^T, 50x50 over F=40) computed with full-precision
// V_WMMA_F32_16X16X4_F32 (matches reference f32 numerics); per-block
// in-LDS Cholesky + 2-RHS triangular solves; deterministic sigsq reduction.
//
// WMMA D tiles are stored UNGUARDED into a 64-row padded LDS buffer: the
// padded xn rows are zeroed, so the padded gram region is exactly zero and
// no exec-mask toggling is needed around the ds_store stream.

typedef __attribute__((ext_vector_type(2))) float v2f;
typedef __attribute__((ext_vector_type(8))) float v8f;

#define NNB   50     // neighbors per batch element
#define FDIM  40     // feature dim
#define MCH   4      // channels / nus
#define XS    44     // LDS row stride for xn (floats): 16B aligned, bank-clean
#define KS    51     // row stride for Cholesky matrix
#define KSP   68     // row stride for padded gram/pair matrix (store-conflict-free)
#define NPAD  64     // neighbor count padded to 4 WMMA tiles of 16
#define TPB   128    // 4 wave32 waves per block
#define EPSI  1e-5f

__device__ __forceinline__ float matern_ch(float d, int ch, float ls) {
  const float r = d / ls;
  if (ch == 1) {            // nu = 1.5
    const float s = 1.7320508075688772f * r;
    return (1.0f + s) * expf(-s);
  }
  if (ch == 2) {            // nu = 2.5
    const float s = 2.23606797749979f * r;
    return (1.0f + s + s * s * (1.0f / 3.0f)) * expf(-s);
  }
  return expf(-r);          // nu = 0.5 (channels 0 and 3)
}

__global__ __launch_bounds__(TPB) void muygps_main(
    const float* __restrict__ x,
    const float* __restrict__ length_scale,
    const float* __restrict__ tgt,            // (B, NN, M)
    const int*   __restrict__ batch_inds,     // (B,)
    const int*   __restrict__ batch_nn_inds,  // (B, NN)
    float* __restrict__ out,                  // preds | varis | sigsq
    float* __restrict__ ws,                   // (M, B) sigsq partials
    int B) {
  __shared__ float xnS[NPAD * XS];    // gathered neighbor rows (padded, zeroed)
  __shared__ float PM[NPAD * KSP];    // gram -> pairwise distances (64-row padded)
  __shared__ float KM[NNB * KS];      // kernel matrix / Cholesky factor
  __shared__ float xbS[FDIM];
  __shared__ float crossS[NNB], sqS[NNB], KcS[NNB], tS[NNB], y0S[NNB], y1S[NNB];
  __shared__ float lsS[MCH];

  const int b    = blockIdx.x;
  const int tid  = threadIdx.x;
  const int wave = tid >> 5;
  const int lane = tid & 31;
  const int l    = lane & 15;   // row-of-A / col-of-B / N index
  const int h    = lane >> 4;   // K half select

  // ---- Phase 1: gather rows into LDS ------------------------------------
  if (tid < MCH) lsS[tid] = length_scale[tid];
  {
    const int bi = batch_inds[b];
    if (tid < FDIM) xbS[tid] = x[(size_t)bi * FDIM + tid];
  }
  for (int idx = tid; idx < (NPAD - NNB) * XS; idx += TPB)
    xnS[NNB * XS + idx] = 0.0f;                       // zero pad rows 50..63
  for (int idx = tid; idx < NNB * (FDIM / 4); idx += TPB) {
    const int row = idx / (FDIM / 4);
    const int q   = idx % (FDIM / 4);
    const int src = batch_nn_inds[b * NNB + row];
    const float4 v = *reinterpret_cast<const float4*>(x + (size_t)src * FDIM + q * 4);
    *reinterpret_cast<float4*>(&xnS[row * XS + q * 4]) = v;  // 16B aligned (XS=44)
  }
  __syncthreads();

  // ---- Phase 2: gram = xn · xn^T via V_WMMA_F32_16X16X4_F32 -------------
  // A(m,k): lane = m%16 + 16*(k>=2), vgpr = k&1 ; B(k,n) mirrors A.
  // D(m,n): vgpr r -> m = i0 + r + 8h, n = j0 + l.
  const int i0 = wave * 16;                           // wave owns one tile row
  const float* arow = &xnS[(i0 + l) * XS + 2 * h];
  for (int tj = 0; tj < 4; ++tj) {
    const int j0 = tj * 16;
    const float* brow = &xnS[(j0 + l) * XS + 2 * h];
    v8f acc = {};
#pragma unroll
    for (int k0 = 0; k0 < FDIM; k0 += 4) {
      const v2f a  = *reinterpret_cast<const v2f*>(arow + k0);
      const v2f bb = *reinterpret_cast<const v2f*>(brow + k0);
      acc = __builtin_amdgcn_wmma_f32_16x16x4_f32(false, a, false, bb,
                                                  (short)0, acc, false, false);
    }
    // Unguarded writeback into padded buffer (padded region is exact zero).
    float* drow = &PM[(i0 + 8 * h) * KSP + j0 + l];
#pragma unroll
    for (int r = 0; r < 8; ++r) drow[r * KSP] = acc[r];
  }
  __syncthreads();

  // cross distances + gram diagonal
  if (tid < NNB) {
    float d2 = 0.0f;
#pragma unroll 8
    for (int k = 0; k < FDIM; ++k) {
      const float d = xbS[k] - xnS[tid * XS + k];
      d2 += d * d;
    }
    crossS[tid] = (d2 > 0.0f) ? sqrtf(d2) : 0.0f;
    sqS[tid]    = PM[tid * KSP + tid];
  }
  __syncthreads();

  // gram -> pairwise distances, in place (50x50 live region only)
  for (int idx = tid; idx < NNB * NNB; idx += TPB) {
    const int i = idx / NNB, j = idx % NNB;
    const float g  = PM[i * KSP + j];
    float d2 = fmaxf(sqS[i] + sqS[j] - 2.0f * g, 0.0f);
    PM[i * KSP + j] = (d2 > 0.0f) ? sqrtf(d2) : 0.0f;
  }
  __syncthreads();

  // ---- Phase 3: per-channel Matern kernel + Cholesky + solves ------------
  for (int ch = 0; ch < MCH; ++ch) {
    const float ls = lsS[ch];
    for (int idx = tid; idx < NNB * NNB; idx += TPB) {
      const int i = idx / NNB, j = idx % NNB;
      float v = matern_ch(PM[i * KSP + j], ch, ls);
      if (i == j) v += EPSI;
      KM[i * KS + j] = v;
    }
    if (tid < NNB) {
      const float kc = matern_ch(crossS[tid], ch, ls);
      const float tv = tgt[((size_t)b * NNB + tid) * MCH + ch];
      KcS[tid] = kc;  tS[tid] = tv;  y0S[tid] = tv;  y1S[tid] = kc;
    }
    __syncthreads();

    // Cholesky (lower, in place), right-looking
    for (int k = 0; k < NNB; ++k) {
      if (tid == 0) KM[k * KS + k] = sqrtf(KM[k * KS + k]);
      __syncthreads();
      const float dinv = 1.0f / KM[k * KS + k];
      for (int i = k + 1 + tid; i < NNB; i += TPB) KM[i * KS + k] *= dinv;
      __syncthreads();
      const int rem = NNB - 1 - k;
      for (int idx = tid; idx < rem * rem; idx += TPB) {
        const int jj = idx / rem, ii = idx % rem;
        const int j = k + 1 + jj, i = k + 1 + ii;
        if (i >= j) KM[i * KS + j] -= KM[i * KS + k] * KM[j * KS + k];
      }
      __syncthreads();
    }

    // Forward solve L y = rhs (two RHS)
    for (int j = 0; j < NNB; ++j) {
      if (tid == 0) y0S[j] /= KM[j * KS + j];
      if (tid == 1) y1S[j] /= KM[j * KS + j];
      __syncthreads();
      const float a0 = y0S[j], a1 = y1S[j];
      for (int i = j + 1 + tid; i < NNB; i += TPB) {
        const float lij = KM[i * KS + j];
        y0S[i] -= lij * a0;
        y1S[i] -= lij * a1;
      }
      __syncthreads();
    }
    // Back solve L^T x = y
    for (int j = NNB - 1; j >= 0; --j) {
      if (tid == 0) y0S[j] /= KM[j * KS + j];
      if (tid == 1) y1S[j] /= KM[j * KS + j];
      __syncthreads();
      const float a0 = y0S[j], a1 = y1S[j];
      for (int i = tid; i < j; i += TPB) {
        const float lji = KM[j * KS + i];
        y0S[i] -= lji * a0;
        y1S[i] -= lji * a1;
      }
      __syncthreads();
    }

    // Outputs: preds, varis, sigsq partial (fixed-order serial dots)
    if (tid == 0) {
      float p = 0.0f;
      for (int j = 0; j < NNB; ++j) p += KcS[j] * y0S[j];
      out[(size_t)b * MCH + ch] = p;
    } else if (tid == 32) {
      float v = 0.0f;
      for (int j = 0; j < NNB; ++j) v += KcS[j] * y1S[j];
      out[(size_t)B * MCH + (size_t)b * MCH + ch] = 1.0f - v;
    } else if (tid == 64) {
      float s = 0.0f;
      for (int j = 0; j < NNB; ++j) s += tS[j] * y0S[j];
      ws[(size_t)ch * B + b] = s;
    }
    __syncthreads();
  }
}

// Deterministic sigsq reduction: fixed strided order + fixed tree order.
__global__ __launch_bounds__(256) void sigsq_reduce(const float* __restrict__ ws,
                                                    float* __restrict__ out, int B) {
  __shared__ float red[256];
  const int ch = blockIdx.x, tid = threadIdx.x;
  float s = 0.0f;
  for (int i = tid; i < B; i += 256) s += ws[(size_t)ch * B + i];
  red[tid] = s;
  __syncthreads();
  for (int off = 128; off > 0; off >>= 1) {
    if (tid < off) red[tid] += red[tid + off];
    __syncthreads();
  }
  if (tid == 0) out[(size_t)2 * B * MCH + ch] = red[0] / (float)((size_t)NNB * B);
}

extern "C" void kernel_launch(void* const* d_in, const int* in_sizes, int n_in,
                              void* d_out, int out_size, void* d_ws, size_t ws_size,
                              hipStream_t stream) {
  (void)n_in; (void)out_size; (void)ws_size;
  const float* x     = (const float*)d_in[0];  // (N, F)
  const float* ls    = (const float*)d_in[1];  // (4,)
  const float* tgt   = (const float*)d_in[2];  // (B, NN, M)
  const int*   binds = (const int*)d_in[3];    // (B,)
  const int*   bnn   = (const int*)d_in[4];    // (B, NN)
  float* out = (float*)d_out;
  float* ws  = (float*)d_ws;
  const int B = in_sizes[3];

  muygps_main<<<dim3(B), dim3(TPB), 0, stream>>>(x, ls, tgt, binds, bnn, out, ws, B);
  sigsq_reduce<<<dim3(MCH), dim3(256), 0, stream>>>(ws, out, B);
}